// SinkhornDistance_26903675142567
// MI455X (gfx1250) — compile-verified
//
#include <hip/hip_runtime.h>

#define NP 64
#define PP 1024
constexpr float EPSF     = 0.1f;
constexpr float THRESHF  = 0.1f;
constexpr float LOG_EPSV = -18.420680744f;      // log(1e-8)
constexpr float LOG1P_EPS = 1.0e-8f;            // log(1+1e-8)
constexpr float K2E      = 14.426950408889634f; // (1/eps) * log2(e)

typedef __attribute__((ext_vector_type(2))) float v2f;
typedef __attribute__((ext_vector_type(4))) float v4f;
typedef __attribute__((ext_vector_type(8))) float v8f;

__device__ __forceinline__ float fast_exp2(float x) {
    return __builtin_amdgcn_exp2f(x);           // raw v_exp_f32
}

// ---------------------------------------------------------------- init ------
__global__ __launch_bounds__(256)
void sink_init(const float* __restrict__ logits,
               float* __restrict__ u, float* __restrict__ v,
               float* __restrict__ log_mu, int* __restrict__ done)
{
    int n = blockIdx.x;
    int tid = threadIdx.x;
    __shared__ float red[256];
    const float* row = logits + (size_t)n * PP;

    float mx = -3.0e38f;
    for (int j = tid; j < PP; j += 256) mx = fmaxf(mx, row[j]);
    red[tid] = mx; __syncthreads();
    for (int s = 128; s > 0; s >>= 1) {
        if (tid < s) red[tid] = fmaxf(red[tid], red[tid + s]);
        __syncthreads();
    }
    mx = red[0]; __syncthreads();

    float sm = 0.0f;
    for (int j = tid; j < PP; j += 256) sm += __expf(row[j] - mx);
    red[tid] = sm; __syncthreads();
    for (int s = 128; s > 0; s >>= 1) {
        if (tid < s) red[tid] += red[tid + s];
        __syncthreads();
    }
    float lse = mx + __logf(red[0]);

    for (int j = tid; j < PP; j += 256) {
        // log(softmax + 1e-8), exactly as the reference
        log_mu[(size_t)n * PP + j] = __logf(__expf(row[j] - lse) + 1.0e-8f);
        u[(size_t)n * PP + j] = 0.0f;
        v[(size_t)n * PP + j] = 0.0f;
    }
    if (n == 0 && tid == 0) *done = 0;
}

// ------------------------------------------------------- sinkhorn half-step -
// mode 0: dst=u, src=v, uses log_mu, writes errPartial
// mode 1: dst=v, src=u, log_nu from class_ids
__global__ __launch_bounds__(128)
void sink_update(const float* __restrict__ C,
                 const float* __restrict__ srcPot,
                 float* __restrict__ dstPot,
                 const float* __restrict__ log_mu,
                 const int* __restrict__ cid,
                 float* __restrict__ errPartial,
                 const int* __restrict__ done,
                 int mode)
{
    if (*done) return;                       // frozen after convergence

    const int n     = blockIdx.y;
    const int chunk = blockIdx.x;            // 16 chunks of 64 rows
    const int wave  = threadIdx.x >> 5;      // 4 waves, 16 rows each
    const int lane  = threadIdx.x & 31;
    const int r16   = lane & 15;
    const int kh    = lane >> 4;             // half-wave id
    const int i0    = chunk * 64 + wave * 16;
    const int row   = i0 + r16;

    __shared__ float s_src[PP];
    for (int t = threadIdx.x; t < PP; t += 128)
        s_src[t] = srcPot[(size_t)n * PP + t];
    __syncthreads();

    const float* Crow = C + (size_t)row * PP;
    const int joff = kh * 4;                 // each lane owns 4 contiguous cols

    // ---- pass 1: per-row max of (src_j - C[row][j]), float4 loads ------
    float m = -3.0e38f;
    for (int j = 0; j < PP; j += 8) {
        v4f c4 = *(const v4f*)(Crow + j + joff);
        v4f s4 = *(const v4f*)(s_src + j + joff);
        m = fmaxf(m, fmaxf(fmaxf(s4.x - c4.x, s4.y - c4.y),
                           fmaxf(s4.z - c4.z, s4.w - c4.w)));
    }
    m = fmaxf(m, __shfl_xor(m, 16, 32));     // both half-rows -> full row max

    // ---- pass 2: row sums of exp2((src_j - C)*K2E - m*K2E) via WMMA ----
    // B = ones => K-slot assignment is irrelevant (sum is commutative), so a
    // lane's contiguous float4 feeds elems 0,1 -> WMMA d0 and 2,3 -> WMMA d1.
    v8f d0 = {0.f,0.f,0.f,0.f,0.f,0.f,0.f,0.f};
    v8f d1 = {0.f,0.f,0.f,0.f,0.f,0.f,0.f,0.f};
    v2f ones; ones.x = 1.0f; ones.y = 1.0f;
    const float mk = m * K2E;

    for (int j = 0; j < PP; j += 8) {
        v4f c4 = *(const v4f*)(Crow + j + joff);
        v4f s4 = *(const v4f*)(s_src + j + joff);
        v2f a, b;
        a.x = fast_exp2((s4.x - c4.x) * K2E - mk);
        a.y = fast_exp2((s4.y - c4.y) * K2E - mk);
        b.x = fast_exp2((s4.z - c4.z) * K2E - mk);
        b.y = fast_exp2((s4.w - c4.w) * K2E - mk);
        d0 = __builtin_amdgcn_wmma_f32_16x16x4_f32(false, a, false, ones,
                                                   (short)0, d0, false, false);
        d1 = __builtin_amdgcn_wmma_f32_16x16x4_f32(false, b, false, ones,
                                                   (short)0, d1, false, false);
    }
    v8f ds = d0 + d1;

    // D layout: lanes 0-15 -> rows i0+0..7 in ds[0..7]; lanes 16-31 -> rows i0+8..15
    int rsel = r16 & 7;
    float Sown = ds[0];
    if (rsel == 1) Sown = ds[1];
    if (rsel == 2) Sown = ds[2];
    if (rsel == 3) Sown = ds[3];
    if (rsel == 4) Sown = ds[4];
    if (rsel == 5) Sown = ds[5];
    if (rsel == 6) Sown = ds[6];
    if (rsel == 7) Sown = ds[7];
    float Sx = __shfl_xor(Sown, 16, 32);
    bool ownValid = (kh == 0) ? (r16 < 8) : (r16 >= 8);
    float S = ownValid ? Sown : Sx;          // this lane's row sum

    // u_new = eps*log_mu - m - eps*log(S)   (the old potential cancels)
    float lm;
    if (mode == 0) lm = log_mu[(size_t)n * PP + row];
    else           lm = (row == cid[n]) ? LOG1P_EPS : LOG_EPSV;
    float pnew = EPSF * lm - m - EPSF * __logf(S);

    size_t idx = (size_t)n * PP + row;
    float du = 0.0f;
    if (mode == 0) {
        float old = dstPot[idx];
        du = fabsf(pnew - old);
    }
    if (kh == 0) dstPot[idx] = pnew;         // one writer per row
    if (kh != 0) du = 0.0f;

    // deterministic |du| reduction: wave tree + 4-wave LDS combine
    for (int off = 16; off > 0; off >>= 1) du += __shfl_xor(du, off, 32);
    if (mode == 0) {
        __shared__ float s_err[4];
        if (lane == 0) s_err[wave] = du;
        __syncthreads();
        if (threadIdx.x == 0)
            errPartial[n * 16 + chunk] = s_err[0] + s_err[1] + s_err[2] + s_err[3];
    }
}

// -------------------------------------------------------- convergence ------
__global__ __launch_bounds__(256)
void sink_check(const float* __restrict__ errPartial, int* __restrict__ done)
{
    if (*done) return;
    int tid = threadIdx.x;
    __shared__ float red[256];
    float s = 0.0f;
    for (int i = tid; i < NP * 16; i += 256) s += errPartial[i];
    red[tid] = s; __syncthreads();
    for (int st = 128; st > 0; st >>= 1) {
        if (tid < st) red[tid] += red[tid + st];
        __syncthreads();
    }
    if (tid == 0 && red[0] * (1.0f / (float)NP) < THRESHF) *done = 1;
}

// ------------------------------------------------------------ final cost ----
__global__ __launch_bounds__(256)
void sink_cost(const float* __restrict__ C,
               const float* __restrict__ u, const float* __restrict__ v,
               float* __restrict__ costPartial)
{
    int n = blockIdx.y, chunk = blockIdx.x, tid = threadIdx.x;
    __shared__ float s_v[PP];
    for (int t = tid; t < PP; t += 256) s_v[t] = v[(size_t)n * PP + t];
    __syncthreads();

    float acc = 0.0f;
    for (int rr = 0; rr < 64; ++rr) {
        int row = chunk * 64 + rr;
        float ur = u[(size_t)n * PP + row];
        const float* Crow = C + (size_t)row * PP;
        v4f c4 = *(const v4f*)(Crow + tid * 4);        // 256 threads * 4 = 1024
        v4f v4 = *(const v4f*)(s_v + tid * 4);
        acc += fast_exp2((ur + v4.x - c4.x) * K2E) * c4.x;
        acc += fast_exp2((ur + v4.y - c4.y) * K2E) * c4.y;
        acc += fast_exp2((ur + v4.z - c4.z) * K2E) * c4.z;
        acc += fast_exp2((ur + v4.w - c4.w) * K2E) * c4.w;
    }
    __shared__ float red[256];
    red[tid] = acc; __syncthreads();
    for (int st = 128; st > 0; st >>= 1) {
        if (tid < st) red[tid] += red[tid + st];
        __syncthreads();
    }
    if (tid == 0) costPartial[n * 16 + chunk] = red[0];
}

__global__ __launch_bounds__(64)
void sink_final(const float* __restrict__ costPartial, float* __restrict__ out)
{
    int n = threadIdx.x;
    float s = 0.0f;
    for (int c = 0; c < 16; ++c) s += costPartial[n * 16 + c];
    out[n] = s;
}

// ---------------------------------------------------------------------------
extern "C" void kernel_launch(void* const* d_in, const int* in_sizes, int n_in,
                              void* d_out, int out_size, void* d_ws, size_t ws_size,
                              hipStream_t stream)
{
    const float* logits = (const float*)d_in[0];
    const int*   cid    = (const int*)d_in[1];
    const float* C      = (const float*)d_in[2];
    float* out = (float*)d_out;

    float* ws   = (float*)d_ws;
    float* u    = ws;                           // 64*1024
    float* v    = ws + 65536;                   // 64*1024
    float* lmu  = ws + 131072;                  // 64*1024
    float* errP = ws + 196608;                  // 64*16
    float* cstP = ws + 197632;                  // 64*16
    int*   done = (int*)(ws + 198656);

    sink_init<<<NP, 256, 0, stream>>>(logits, u, v, lmu, done);

    dim3 grid(16, NP);
    for (int it = 0; it < 100; ++it) {
        sink_update<<<grid, 128, 0, stream>>>(C, v, u, lmu, cid, errP, done, 0);
        sink_update<<<grid, 128, 0, stream>>>(C, u, v, lmu, cid, errP, done, 1);
        sink_check <<<1, 256, 0, stream>>>(errP, done);
    }
    sink_cost <<<grid, 256, 0, stream>>>(C, u, v, cstP);
    sink_final<<<1, 64, 0, stream>>>(cstP, out);
}